// MonarchFFN_24558622998777
// MI455X (gfx1250) — compile-verified
//
#include <hip/hip_runtime.h>
#include <hip/hip_bf16.h>

typedef __attribute__((ext_vector_type(2))) float v2f;
typedef __attribute__((ext_vector_type(8))) float v8f;

#define SDIM     32
#define DDIM     1024
#define TROWS    32          // rows per workgroup (2 row-groups of 16 -> B-fragment reuse)
#define LSTR     1028        // padded LDS row stride (floats): 1028%64==4 -> conflict-free strided gathers, 16B aligned rows
#define NWAVES   8
#define NTHREADS 256

// ---- B fragment: 4x16 fp32, mirrored A layout: VGPR0 = K=koff row, VGPR1 = K=koff+1; lanes>=16 use K+2 ----
__device__ __forceinline__ void load_B(const float* __restrict__ Wb, int l16, int koff, v2f Bf[8][2]) {
#pragma unroll
    for (int s = 0; s < 8; ++s) {
        int kb = s * 4 + koff;
#pragma unroll
        for (int h = 0; h < 2; ++h) {
            const float* p = Wb + (h * 16 + l16) * SDIM + kb;  // B[k][n] = W[b][n][k]; k contiguous -> b64 load
            v2f t; t.x = p[0]; t.y = p[1];
            Bf[s][h] = t;
        }
    }
}

__device__ __forceinline__ void wmma_chain(const v2f* Af, const v2f (*Bf)[2], v8f& acc0, v8f& acc1) {
#pragma unroll
    for (int s = 0; s < 8; ++s) {
        acc0 = __builtin_amdgcn_wmma_f32_16x16x4_f32(false, Af[s], false, Bf[s][0], (short)0, acc0, false, false);
        acc1 = __builtin_amdgcn_wmma_f32_16x16x4_f32(false, Af[s], false, Bf[s][1], (short)0, acc1, false, false);
    }
}

// store D (16x16 pair) to LDS in natural layout out[t][b*32+n]; conflict-free
__device__ __forceinline__ void store_D(float* __restrict__ outbuf, int g, int b, int l16, int lhi,
                                        const v8f& acc0, const v8f& acc1) {
#pragma unroll
    for (int r = 0; r < 8; ++r) {
        int t = g * 16 + r + 8 * lhi;
        outbuf[t * LSTR + b * SDIM + l16]      = acc0[r];
        outbuf[t * LSTR + b * SDIM + 16 + l16] = acc1[r];
    }
}

// Stages 1/2: A[t][m] = in[t][m*32 + b]   (perm folded in); B reused across both row-groups
__device__ __forceinline__ void stage_strided(const float* __restrict__ inbuf, float* __restrict__ outbuf,
                                              const float* __restrict__ W, int wave, int lane) {
    const int l16 = lane & 15, lhi = lane >> 4, koff = lhi * 2;
    for (int bi = 0; bi < 4; ++bi) {
        const int b = wave + bi * NWAVES;
        v2f Bf[8][2];
        load_B(W + b * SDIM * SDIM, l16, koff, Bf);
#pragma unroll
        for (int g = 0; g < 2; ++g) {
            v2f Af[8];
            const float* inrow = inbuf + (g * 16 + l16) * LSTR + b;
#pragma unroll
            for (int s = 0; s < 8; ++s) {
                int m0 = s * 4 + koff;
                v2f a; a.x = inrow[m0 * SDIM]; a.y = inrow[(m0 + 1) * SDIM];
                Af[s] = a;
            }
            v8f acc0 = {}, acc1 = {};
            wmma_chain(Af, Bf, acc0, acc1);
            store_D(outbuf, g, b, l16, lhi, acc0, acc1);
        }
    }
}

// Stage 3: A[t][m] = relu(dk[m*32+b] * v[t][b*32+m]) + b1[m*32+b]  (monarch1's trailing perm folded in)
__device__ __forceinline__ void stage_ewise(const float* __restrict__ inbuf, float* __restrict__ outbuf,
                                            const float* __restrict__ W, const float* __restrict__ dk,
                                            const float* __restrict__ b1v, int wave, int lane) {
    const int l16 = lane & 15, lhi = lane >> 4, koff = lhi * 2;
    for (int bi = 0; bi < 4; ++bi) {
        const int b = wave + bi * NWAVES;
        v2f Bf[8][2];
        load_B(W + b * SDIM * SDIM, l16, koff, Bf);
        // hoist per-(m,b) elementwise constants: independent of row-group
        v2f Ek[8], Ck[8];
#pragma unroll
        for (int s = 0; s < 8; ++s) {
            int m0 = s * 4 + koff;
            v2f e; e.x = dk[m0 * SDIM + b];  e.y = dk[(m0 + 1) * SDIM + b];  Ek[s] = e;
            v2f c; c.x = b1v[m0 * SDIM + b]; c.y = b1v[(m0 + 1) * SDIM + b]; Ck[s] = c;
        }
#pragma unroll
        for (int g = 0; g < 2; ++g) {
            v2f Af[8];
            const float* vrow = inbuf + (g * 16 + l16) * LSTR + b * SDIM;  // contiguous in m
#pragma unroll
            for (int s = 0; s < 8; ++s) {
                int m0 = s * 4 + koff;
                v2f a;
                a.x = fmaxf(Ek[s].x * vrow[m0],     0.0f) + Ck[s].x;
                a.y = fmaxf(Ek[s].y * vrow[m0 + 1], 0.0f) + Ck[s].y;
                Af[s] = a;
            }
            v8f acc0 = {}, acc1 = {};
            wmma_chain(Af, Bf, acc0, acc1);
            store_D(outbuf, g, b, l16, lhi, acc0, acc1);
        }
    }
}

// Stage 4: acc initialized with b2 (out layout), result scattered to global with final perm folded in:
//   out[row][n*32 + b] = v2[row][b*32+n] + b2[n*32+b]
__device__ __forceinline__ void stage_final(const float* __restrict__ inbuf, float* __restrict__ outg,
                                            const float* __restrict__ W, const float* __restrict__ b2v,
                                            int wave, int lane) {
    const int l16 = lane & 15, lhi = lane >> 4, koff = lhi * 2;
    for (int bi = 0; bi < 4; ++bi) {
        const int b = wave + bi * NWAVES;
        v2f Bf[8][2];
        load_B(W + b * SDIM * SDIM, l16, koff, Bf);
        const float c0 = b2v[l16 * SDIM + b];          // n = l16,     same for all 8 fragment rows
        const float c1 = b2v[(l16 + 16) * SDIM + b];   // n = l16+16
#pragma unroll
        for (int g = 0; g < 2; ++g) {
            v2f Af[8];
            const float* inrow = inbuf + (g * 16 + l16) * LSTR + b;
#pragma unroll
            for (int s = 0; s < 8; ++s) {
                int m0 = s * 4 + koff;
                v2f a; a.x = inrow[m0 * SDIM]; a.y = inrow[(m0 + 1) * SDIM];
                Af[s] = a;
            }
            v8f acc0 = {c0, c0, c0, c0, c0, c0, c0, c0};
            v8f acc1 = {c1, c1, c1, c1, c1, c1, c1, c1};
            wmma_chain(Af, Bf, acc0, acc1);
#pragma unroll
            for (int r = 0; r < 8; ++r) {
                int t = g * 16 + r + 8 * lhi;
                outg[(size_t)t * DDIM + l16 * SDIM + b]        = acc0[r];
                outg[(size_t)t * DDIM + (l16 + 16) * SDIM + b] = acc1[r];
            }
        }
    }
}

__global__ __launch_bounds__(NTHREADS, 1)
void MonarchFFN_fused_wmma(const float* __restrict__ x,
                           const float* __restrict__ L1, const float* __restrict__ R1,
                           const float* __restrict__ L2, const float* __restrict__ R2,
                           const float* __restrict__ dk, const float* __restrict__ b1v,
                           const float* __restrict__ b2v, float* __restrict__ out) {
    __shared__ __align__(16) float buf0[TROWS * LSTR];
    __shared__ __align__(16) float buf1[TROWS * LSTR];

    const int tid  = threadIdx.x;
    const int wave = tid >> 5;
    const int lane = tid & 31;
    const size_t row0 = (size_t)blockIdx.x * TROWS;

    // cooperative, coalesced float4 load of the 32-row x tile into padded LDS
    {
        const float4* xg = (const float4*)(x + row0 * DDIM);
#pragma unroll
        for (int i = 0; i < (TROWS * DDIM / 4) / NTHREADS; ++i) {
            int f  = tid + i * NTHREADS;     // float4 slots
            int r  = f >> 8;                 // row (256 float4 per row)
            int c4 = f & 255;
            *(float4*)&buf0[r * LSTR + c4 * 4] = xg[r * 256 + c4];
        }
    }
    __syncthreads();

    stage_strided(buf0, buf1, L1, wave, lane);            // u  = S(x, L1)
    __syncthreads();
    stage_strided(buf1, buf0, R1, wave, lane);            // v  = S(u, R1)
    __syncthreads();
    stage_ewise(buf0, buf1, L2, dk, b1v, wave, lane);     // u2 = S(relu(dk*P(v))+b1, L2)
    __syncthreads();
    stage_final(buf1, out + row0 * DDIM, R2, b2v, wave, lane);  // out = P(S(u2,R2)) + b2
}

extern "C" void kernel_launch(void* const* d_in, const int* in_sizes, int n_in,
                              void* d_out, int out_size, void* d_ws, size_t ws_size,
                              hipStream_t stream) {
    const float* x  = (const float*)d_in[0];
    const float* L1 = (const float*)d_in[1];
    const float* R1 = (const float*)d_in[2];
    const float* L2 = (const float*)d_in[3];
    const float* R2 = (const float*)d_in[4];
    const float* dk = (const float*)d_in[5];
    const float* b1 = (const float*)d_in[6];
    const float* b2 = (const float*)d_in[7];
    float* out = (float*)d_out;

    const int rows = in_sizes[0] / DDIM;   // 8*4096 = 32768
    const int wgs  = rows / TROWS;         // 1024
    hipLaunchKernelGGL(MonarchFFN_fused_wmma, dim3(wgs), dim3(NTHREADS), 0, stream,
                       x, L1, R1, L2, R2, dk, b1, b2, out);
}